// LRAP_loss_42691974922893
// MI455X (gfx1250) — compile-verified
//
#include <hip/hip_runtime.h>
#include <stdint.h>

// LRAP loss: B=16384 rows, C=2048 classes.
// Per row (one 256-thread / 8-wave block):
//  - build 64-bit keys [~ord(pred):32][idx:11][label:1] in registers (8/thread)
//  - bitonic sort: shfl_xor for j<=16, register swaps for j=32/64/128,
//    LDS only for the 6 substages with j>=256
//  - WMMA matmul-scan (lower-triangular ones @ labels) for positive prefix-counts
//  - accumulate  lab[p] * cum[p] / (p+1),  / n_pos, mean over rows.

#define B_ROWS 16384
#define C_CLS  2048
#define THREADS 256
#define EPT     8            // elements per thread
#define NSEG   (C_CLS / 16)  // 128 segments of 16

typedef unsigned long long u64;
typedef __attribute__((ext_vector_type(2))) float v2f;
typedef __attribute__((ext_vector_type(8))) float v8f;

__global__ void lrap_zero_out(float* out) { out[0] = 0.0f; }

// bitonic compare-exchange: element e holds a, partner (e^j) holds b.
// ascending region iff (e & k)==0; lower element of the pair keeps min.
__device__ __forceinline__ u64 pick_cex(u64 a, u64 b, unsigned e, unsigned j, unsigned k) {
    const bool up      = ((e & k) == 0);
    const bool lower   = ((e & j) == 0);
    const bool keepmin = (lower == up);
    const bool less    = (a < b);
    return (less == keepmin) ? a : b;
}

__global__ __launch_bounds__(THREADS)
void lrap_kernel(const float* __restrict__ preds,
                 const float* __restrict__ labels,
                 float* __restrict__ out)
{
    __shared__ u64   keys[C_CLS];    // 16 KB, used only for j>=256 exchanges
    __shared__ float labf[C_CLS];    // 8 KB: sorted labels as f32
    __shared__ float segoff[NSEG];   // totals -> exclusive offsets (in place)
    __shared__ float npos_s;
    __shared__ float blocksum;

    const int row  = blockIdx.x;
    const int t    = threadIdx.x;
    const int wave = t >> 5;                       // 0..7
    const int lane = t & 31;
    const unsigned ebase = (unsigned)(wave * 256 + lane);  // + r*32, r=0..7

    if (t == 0) blocksum = 0.0f;

    const float* prow = preds  + (size_t)row * C_CLS;
    const float* lrow = labels + (size_t)row * C_CLS;

    // ---- build keys in registers ----
    u64 key[EPT];
    #pragma unroll
    for (int r = 0; r < EPT; ++r) {
        const unsigned e = ebase + (unsigned)r * 32u;
        unsigned u = __float_as_uint(prow[e]);
        u ^= (u >> 31) ? 0xFFFFFFFFu : 0x80000000u;   // monotonic float->uint
        const unsigned lab = (lrow[e] != 0.0f) ? 1u : 0u;
        key[r] = ((u64)(~u) << 12) | ((u64)e << 1) | (u64)lab;
    }

    // ---- bitonic sort ascending, register-resident ----
    for (unsigned k = 2; k <= (unsigned)C_CLS; k <<= 1) {
        // cross-wave exchanges (j >= 256): via LDS (6 substages total)
        for (unsigned j = k >> 1; j >= 256; j >>= 1) {
            __syncthreads();                  // previous substage's reads done
            #pragma unroll
            for (int r = 0; r < EPT; ++r) keys[ebase + (unsigned)r * 32u] = key[r];
            __syncthreads();
            #pragma unroll
            for (int r = 0; r < EPT; ++r) {
                const unsigned e = ebase + (unsigned)r * 32u;
                u64 o = keys[e ^ j];
                key[r] = pick_cex(key[r], o, e, j, k);
            }
        }
        // j in {128,64,32}: partner register r ^ (j>>5), thread-local
        #pragma unroll
        for (int m = 4; m >= 1; m >>= 1) {
            const unsigned j = (unsigned)m << 5;
            if (j <= (k >> 1)) {
                #pragma unroll
                for (int r = 0; r < EPT; ++r) {
                    if ((r & m) == 0) {
                        const unsigned e = ebase + (unsigned)r * 32u;
                        const bool up = ((e & k) == 0);
                        u64 a = key[r], b = key[r | m];
                        if ((a > b) == up) { key[r] = b; key[r | m] = a; }
                    }
                }
            }
        }
        // j <= 16: crosslane within the wave, barrier-free
        #pragma unroll
        for (int jb = 4; jb >= 0; --jb) {
            const unsigned j = 1u << jb;
            if (j < k) {                      // j <= k/2 for powers of two
                #pragma unroll
                for (int r = 0; r < EPT; ++r) {
                    u64 o = __shfl_xor(key[r], (int)j, 32);
                    key[r] = pick_cex(key[r], o, ebase + (unsigned)r * 32u, j, k);
                }
            }
        }
    }

    // ---- sorted labels -> f32 in LDS (position = element index after sort) ----
    #pragma unroll
    for (int r = 0; r < EPT; ++r) {
        labf[ebase + (unsigned)r * 32u] = (float)(unsigned)(key[r] & 1ull);
    }
    __syncthreads();

    // ---- WMMA matmul-scan: Y = L(16x16 lower-tri ones) @ X, K=16 via 4x K=4 ----
    // Wave w (0..7) scans tile of 256 elems: X[i][n] = labf[tile*256 + n*16 + i].
    // A (32-bit 16x4): lane m%16 holds row m; vgpr v, half h -> K col = v + 2h.
    // C/D (16x16 f32): vgpr v, half h -> M = v + 8h, N = lane%16.
    const int half  = lane >> 4;
    const int lan16 = lane & 15;
    const int tile  = wave;
    v8f acc = {};

    #pragma unroll
    for (int c = 0; c < 4; ++c) {
        v2f a, b;
        #pragma unroll
        for (int v = 0; v < 2; ++v) {
            const int j = 4 * c + v + 2 * half;       // global K index 0..15
            a[v] = (j <= lan16) ? 1.0f : 0.0f;        // lower-triangular ones
            b[v] = labf[tile * 256 + lan16 * 16 + j]; // X[j][n], n = lan16
        }
        acc = __builtin_amdgcn_wmma_f32_16x16x4_f32(
                  false, a, false, b, (short)0, acc, false, false);
    }
    // segment totals: Y[15][n] = vgpr7 of lanes 16..31
    if (half == 1) segoff[tile * 16 + lan16] = acc[7];
    __syncthreads();

    // ---- wave-0 shuffle scan of 128 segment totals -> exclusive offsets ----
    if (wave == 0) {
        const int s = lane * 4;
        float v0 = segoff[s + 0], v1 = segoff[s + 1];
        float v2 = segoff[s + 2], v3 = segoff[s + 3];
        float s0 = v0, s1 = s0 + v1, s2 = s1 + v2, s3 = s2 + v3; // local inclusive
        float incl = s3;                                          // lane total
        #pragma unroll
        for (int d = 1; d < 32; d <<= 1) {
            float n = __shfl_up(incl, d, 32);
            if (lane >= d) incl += n;
        }
        const float excl = incl - s3;
        segoff[s + 0] = excl;
        segoff[s + 1] = excl + s0;
        segoff[s + 2] = excl + s1;
        segoff[s + 3] = excl + s2;
        if (lane == 31) npos_s = incl;   // total positives in this row
    }
    __syncthreads();

    // ---- accumulate terms: lab[p] * cum[p] / (p+1) ----
    {
        float partial = 0.0f;
        const float off = segoff[tile * 16 + lan16];
        #pragma unroll
        for (int v = 0; v < 8; ++v) {
            const int i = v + 8 * half;                 // M index
            const int p = tile * 256 + lan16 * 16 + i;  // sorted position
            partial += labf[p] * (acc[v] + off) / (float)(p + 1);
        }
        // wave-level reduction, one LDS atomic per wave
        #pragma unroll
        for (int d = 16; d >= 1; d >>= 1) partial += __shfl_xor(partial, d, 32);
        if (lane == 0) atomicAdd(&blocksum, partial);
    }
    __syncthreads();

    if (t == 0) {
        atomicAdd(out, blocksum / npos_s * (1.0f / (float)B_ROWS));
    }
}

extern "C" void kernel_launch(void* const* d_in, const int* in_sizes, int n_in,
                              void* d_out, int out_size, void* d_ws, size_t ws_size,
                              hipStream_t stream) {
    const float* preds  = (const float*)d_in[0];
    const float* labels = (const float*)d_in[1];
    float* out = (float*)d_out;

    lrap_zero_out<<<1, 1, 0, stream>>>(out);
    lrap_kernel<<<B_ROWS, THREADS, 0, stream>>>(preds, labels, out);
}